// NALU_14010183319875
// MI455X (gfx1250) — compile-verified
//
#include <hip/hip_runtime.h>

// ---------------------------------------------------------------------------
// NALU forward, MI455X (gfx1250, wave32).
// All three paths expressed as f32 WMMA GEMMs (V_WMMA_F32_16X16X4_F32):
//   a1 = x @ W1
//   p  = log(max(|x|,eps)) @ W2              -> m1 = exp(min(p, 20))
//   q  = neg @ log|1-2w| + zero @ log(1-w)   (w = |w2[u,d]|)  -> |ms| = exp(q)
//   c  = neg @ [1-2w<0]                      -> sign(ms) = (-1)^c
// out = sig(g)*a1 + (1-sig(g))*m1*clip(ms,-1,1)
//
// Tiling: 16x128 output tile per WG (8 waves, one 16x16 WMMA tile each; all
// waves share the A fragments). The x tile is DMA'd into LDS by the Tensor
// Data Mover with hardware padding to a 132-float row stride, then the
// derived planes (log|x|, neg, zero) are computed ONCE per element into LDS.
// The 32-step k-loop is pure {ds_load_2addr_b64, global_load_b64 (L2-resident
// weights), v_wmma_f32_16x16x4_f32} - no data-dependent VALU.
// ---------------------------------------------------------------------------

#define B_SZ   8192
#define D_SZ   128
#define U_SZ   128
#define CLIPV  20.0f
#define EPSV   1e-7f

#define TM     16          // output rows per workgroup
#define LDX    132         // padded LDS row stride in floats
#define PL     (TM*LDX)    // floats per LDS plane
#define WCH    (D_SZ*U_SZ) // floats per weight channel in workspace

typedef __attribute__((ext_vector_type(2))) float v2f;
typedef __attribute__((ext_vector_type(8))) float v8f;
typedef __attribute__((ext_vector_type(4))) unsigned int u32x4;
typedef __attribute__((ext_vector_type(4))) int i32x4;
typedef __attribute__((ext_vector_type(8))) int i32x8;

#if defined(__has_builtin)
#if __has_builtin(__builtin_amdgcn_tensor_load_to_lds) && \
    __has_builtin(__builtin_amdgcn_s_wait_tensorcnt)
#define NALU_HAS_TDM 1
#endif
#endif
#ifndef NALU_HAS_TDM
#define NALU_HAS_TDM 0
#endif

__device__ __forceinline__ float sigf(float v) { return 1.0f / (1.0f + expf(-v)); }

// ---------------------------------------------------------------------------
// Prep: bake the 5 weight channels, stored transposed [n][k] (n=unit, k=d)
// so the GEMM B-fragment reads are contiguous float2's along k.
// ---------------------------------------------------------------------------
__global__ __launch_bounds__(256) void nalu_prep(
    const float* __restrict__ w_hat, const float* __restrict__ m_hat,
    const float* __restrict__ w_hat_p, const float* __restrict__ m_hat_p,
    float* __restrict__ ws) {
  int idx = blockIdx.x * blockDim.x + threadIdx.x;   // 0..16383
  int u = idx >> 7;        // row of transposed store
  int d = idx & 127;       // col (k index)
  int td = d * U_SZ + u;   // transposed read  (element [d,u])
  int dd = u * U_SZ + d;   // direct read      (element [u,d])

  float w1  = tanhf(w_hat[td])   * sigf(m_hat[td]);     // w1[d,u]
  float w2t = tanhf(w_hat_p[td]) * sigf(m_hat_p[td]);   // w2[d,u]
  float w2d = tanhf(w_hat_p[dd]) * sigf(m_hat_p[dd]);   // w2[u,d]

  float wa   = fabsf(w2d);
  float t    = 1.0f - 2.0f * wa;
  float lneg = logf(fmaxf(fabsf(t), 1e-30f));
  float lzer = logf(fmaxf(1.0f - wa, 1e-30f));          // 1-wa >= 0 always
  float nneg = (t < 0.0f) ? 1.0f : 0.0f;

  int o = u * D_SZ + d;
  ws[0 * WCH + o] = w1;
  ws[1 * WCH + o] = w2t;
  ws[2 * WCH + o] = lneg;
  ws[3 * WCH + o] = lzer;
  ws[4 * WCH + o] = nneg;
}

// ---------------------------------------------------------------------------
// Main: 16x128 output tile per WG (8 waves, each one 16x16 WMMA tile).
// ---------------------------------------------------------------------------
__global__ __launch_bounds__(256) void nalu_main(
    const float* __restrict__ x, const float* __restrict__ ws,
    const float* __restrict__ g, float* __restrict__ out) {
  // One shared block: xs at LDS byte offset 0 (used as TDM D#.lds_addr).
  __shared__ float smem[4 * PL];                // 4 * 16 * 132 * 4B = 33 KB
  float* xs  = smem;                            // raw x tile   [16][132]
  float* lxs = smem + 1 * PL;                   // log|x| plane
  float* ngs = smem + 2 * PL;                   // (x<0)  plane
  float* zrs = smem + 3 * PL;                   // (x==0) plane

  const int tid     = threadIdx.x;
  const int rowbase = blockIdx.x * TM;

#if NALU_HAS_TDM
  // --- Tensor Data Mover: DMA 16x128 f32 tile, pad rows 128dw -> +4dw ------
  // D# group0: count=1 | lds_addr | global_addr | type=2   (ISA 8.3)
  // D# group1: data_size=4B, pad_enable, pad_interval=128dw, pad_amount=4dw,
  //            tensor_dim0=128, tensor_dim1=8192, tile=128x16, stride0=128.
  if (tid < 32) {  // one wave issues the DMA (TDM ignores EXEC)
    unsigned long long ga =
        (unsigned long long)(const void*)(x + (size_t)rowbase * D_SZ);
    u32x4 g0 = { 1u,                                   // count=1, user desc
                 0u,                                   // lds_addr = xs @ 0
                 (unsigned)(ga & 0xffffffffu),
                 (unsigned)((ga >> 32) & 0x01ffffffu) | 0x80000000u }; // type=2
    i32x8 g1 = { (int)((2u << 16) | (1u << 20) | (6u << 22) | (3u << 25)),
                 (int)(128u << 16),                    // tensor_dim0 = 128
                 (int)(8192u << 16),                   // tensor_dim1 = 8192
                 (int)(128u << 16),                    // tile_dim0   = 128
                 TM,                                   // tile_dim1   = 16
                 128,                                  // tensor_dim0_stride
                 0, 0 };
    i32x4 zz = { 0, 0, 0, 0 };
#if defined(__clang_major__) && (__clang_major__ >= 23)
    i32x8 z8 = { 0, 0, 0, 0, 0, 0, 0, 0 };
    __builtin_amdgcn_tensor_load_to_lds(g0, g1, zz, zz, z8, 0);
#else
    __builtin_amdgcn_tensor_load_to_lds(g0, g1, zz, zz, 0);
#endif
  }
#endif

  // Warm all 5 weight channels (L2-resident, reused by every WG)
  // -> global_prefetch_b8, 256B granularity covers 64KB per channel.
#pragma unroll
  for (int ch = 0; ch < 5; ++ch)
    __builtin_prefetch(ws + ch * WCH + tid * 64, 0, 1);

#if NALU_HAS_TDM
  if (tid < 32) __builtin_amdgcn_s_wait_tensorcnt(0);
#else
  // Fallback staging: 512 float4's across 256 threads.
#pragma unroll
  for (int i = 0; i < 2; ++i) {
    int f  = tid + i * 256;
    int r  = f >> 5;          // 32 float4 per 128-float row
    int c4 = f & 31;
    float4 v = ((const float4*)(x + (size_t)(rowbase + r) * D_SZ))[c4];
    *(float4*)&xs[r * LDX + c4 * 4] = v;
  }
#endif
  __syncthreads();

  // Derived planes, computed ONCE per element (8-way wave reuse afterwards):
  // log(max(|x|,eps)), neg mask, zero mask. Hoists ALL transcendental and
  // compare work out of the k-loop.
#pragma unroll
  for (int i = 0; i < 8; ++i) {
    int f = tid + i * 256;    // 2048 elements
    int r = f >> 7;
    int c = f & 127;
    int o = r * LDX + c;
    float v = xs[o];
    lxs[o] = logf(fmaxf(fabsf(v), EPSV));
    ngs[o] = (v < 0.0f)  ? 1.0f : 0.0f;
    zrs[o] = (v == 0.0f) ? 1.0f : 0.0f;
  }
  __syncthreads();

  const int lane = tid & 31;
  const int wv   = tid >> 5;       // wave 0..7 = column sub-tile
  const int ml   = lane & 15;
  const int hi   = lane >> 4;      // lanes 16..31 carry K+2,K+3 halves

  const int am = ml;               // A row inside LDS tile (all waves share A)
  const int bn = wv * 16 + ml;     // global output column for B frag (0..127)

  const float* w1p = ws + 0 * WCH + bn * D_SZ;
  const float* w2p = ws + 1 * WCH + bn * D_SZ;
  const float* lnp = ws + 2 * WCH + bn * D_SZ;
  const float* lzp = ws + 3 * WCH + bn * D_SZ;
  const float* nnp = ws + 4 * WCH + bn * D_SZ;

  v8f acc_a1 = {};  // x @ W1
  v8f acc_p  = {};  // logx @ W2
  v8f acc_q  = {};  // neg@Lneg + zero@Lzero
  v8f acc_c  = {};  // neg@Nneg (parity count, integer-exact in f32)

#pragma unroll 8
  for (int k0 = 0; k0 < D_SZ; k0 += 4) {
    const int ka = k0 + hi * 2;   // ISA A/B layout: lo lanes K,K+1; hi K+2,K+3
    const int ao = am * LDX + ka;

    v2f xa = *(const v2f*)&xs[ao];
    v2f lx = *(const v2f*)&lxs[ao];
    v2f ng = *(const v2f*)&ngs[ao];
    v2f zr = *(const v2f*)&zrs[ao];

    v2f b1 = *(const v2f*)(w1p + ka);
    v2f b2 = *(const v2f*)(w2p + ka);
    v2f bl = *(const v2f*)(lnp + ka);
    v2f bz = *(const v2f*)(lzp + ka);
    v2f bc = *(const v2f*)(nnp + ka);

    acc_a1 = __builtin_amdgcn_wmma_f32_16x16x4_f32(false, xa, false, b1, (short)0, acc_a1, false, false);
    acc_p  = __builtin_amdgcn_wmma_f32_16x16x4_f32(false, lx, false, b2, (short)0, acc_p,  false, false);
    acc_q  = __builtin_amdgcn_wmma_f32_16x16x4_f32(false, ng, false, bl, (short)0, acc_q,  false, false);
    acc_q  = __builtin_amdgcn_wmma_f32_16x16x4_f32(false, zr, false, bz, (short)0, acc_q,  false, false);
    acc_c  = __builtin_amdgcn_wmma_f32_16x16x4_f32(false, ng, false, bc, (short)0, acc_c,  false, false);
  }

  // Epilogue. C/D layout: VGPR i -> row i (lanes 0-15) / row i+8 (lanes 16-31),
  // col = lane&15. bn is this lane's output column for every i.
  const float g1 = sigf(g[bn]);
#pragma unroll
  for (int i = 0; i < 8; ++i) {
    int m_out = rowbase + i + hi * 8;
    float a1  = acc_a1[i];
    float m1  = expf(fminf(acc_p[i], CLIPV));
    float msv = expf(acc_q[i]);
    int   cnt = (int)(acc_c[i] + 0.5f);
    if (cnt & 1) msv = -msv;
    msv = fminf(fmaxf(msv, -1.0f), 1.0f);
    out[(size_t)m_out * U_SZ + bn] = g1 * a1 + (1.0f - g1) * m1 * msv;
  }
}

extern "C" void kernel_launch(void* const* d_in, const int* in_sizes, int n_in,
                              void* d_out, int out_size, void* d_ws, size_t ws_size,
                              hipStream_t stream) {
  (void)in_sizes; (void)n_in; (void)out_size; (void)ws_size;
  const float* x   = (const float*)d_in[0];
  const float* wh  = (const float*)d_in[1];
  const float* mh  = (const float*)d_in[2];
  const float* whp = (const float*)d_in[3];
  const float* mhp = (const float*)d_in[4];
  const float* g   = (const float*)d_in[5];
  float* ws  = (float*)d_ws;   // needs 5*128*128*4 = 320 KB
  float* out = (float*)d_out;

  nalu_prep<<<dim3(64), dim3(256), 0, stream>>>(wh, mh, whp, mhp, ws);
  nalu_main<<<dim3(B_SZ / TM), dim3(256), 0, stream>>>(x, ws, g, out);
}